// SpatialAttentionLayer_20761871909585
// MI455X (gfx1250) — compile-verified
//
#include <hip/hip_runtime.h>

typedef __attribute__((ext_vector_type(2))) float v2f;
typedef __attribute__((ext_vector_type(8))) float v8f;

#define DM 128

// Kernel A: r[k] = sum_{n=1..4} relu( colsum(Wn)[k] + bn[k] )
// One block of 256 threads = 8 waves; wave w owns columns [16w, 16w+16).
// Column sum done as ones(16x4) @ W-tile via V_WMMA_F32_16X16X4_F32,
// K accumulated over 32 chained WMMAs (f32 in / f32 acc => full precision).
__global__ __launch_bounds__(256) void colsum_relu_wmma_kernel(
    const float* __restrict__ W1, const float* __restrict__ b1,
    const float* __restrict__ W2, const float* __restrict__ b2,
    const float* __restrict__ W3, const float* __restrict__ b3,
    const float* __restrict__ W4, const float* __restrict__ b4,
    float* __restrict__ r)
{
    const int tid  = threadIdx.x;
    const int wave = tid >> 5;          // 0..7 -> 16-column tile
    const int lane = tid & 31;
    const int col  = wave * 16 + (lane & 15);   // N for this lane
    const int khi  = (lane >> 4) * 2;           // lanes 0-15: K=k0,k0+1 ; 16-31: K=k0+2,k0+3

    const float* Ws[4] = {W1, W2, W3, W4};
    const float* bs[4] = {b1, b2, b3, b4};

    v2f a;                  // A = ones(16x4): every lane holds 1.0 in both A VGPRs
    a.x = 1.0f;
    a.y = 1.0f;

    float total = 0.0f;
#pragma unroll
    for (int m = 0; m < 4; ++m) {
        const float* __restrict__ W = Ws[m];
        v8f acc = {};
#pragma unroll
        for (int k0 = 0; k0 < DM; k0 += 4) {
            v2f bf;
            bf.x = W[(k0 + khi + 0) * DM + col];
            bf.y = W[(k0 + khi + 1) * DM + col];
            // D = A(ones 16x4) * B(4x16 tile) + C  -> rows of D = column sums
            acc = __builtin_amdgcn_wmma_f32_16x16x4_f32(
                false, a, false, bf, (short)0, acc, false, false);
        }
        // D layout: VGPR0 = row M=0 (lanes 0-15, N=lane) / M=8 (lanes 16-31);
        // all rows identical (= column sum), so acc[0] is the column sum on every lane.
        float cs = acc[0];
        total += fmaxf(cs + bs[m][col], 0.0f);
    }
    if (lane < 16) r[col] = total;      // 8 waves x 16 lanes cover all 128 columns
}

// Kernel B: broadcast r[128] over the [8,12,64,128] output, float4-vectorized.
// grid stride is a multiple of 32 float4s, so (i & 31) is loop-invariant per thread.
__global__ __launch_bounds__(256) void broadcast_kernel(
    const float* __restrict__ r, float4* __restrict__ out4, int n4)
{
    int i = blockIdx.x * blockDim.x + threadIdx.x;
    const int stride = gridDim.x * blockDim.x;      // multiple of 256
    const float4* __restrict__ r4 = (const float4*)r;
    float4 rv = r4[i & 31];                          // 128 floats = 32 float4 period
    for (; i < n4; i += stride)
        out4[i] = rv;
}

extern "C" void kernel_launch(void* const* d_in, const int* in_sizes, int n_in,
                              void* d_out, int out_size, void* d_ws, size_t ws_size,
                              hipStream_t stream) {
    // setup_inputs order: 0:X, 1..18: a*/b*/c* attention params (provably unused:
    // softmax over a constant axis is uniform), 19..26: W1,b1,W2,b2,W3,b3,W4,b4
    const float* W1 = (const float*)d_in[19];
    const float* b1 = (const float*)d_in[20];
    const float* W2 = (const float*)d_in[21];
    const float* b2 = (const float*)d_in[22];
    const float* W3 = (const float*)d_in[23];
    const float* b3 = (const float*)d_in[24];
    const float* W4 = (const float*)d_in[25];
    const float* b4 = (const float*)d_in[26];

    float* r = (float*)d_ws;            // 128 floats of scratch, fully overwritten

    colsum_relu_wmma_kernel<<<1, 256, 0, stream>>>(W1, b1, W2, b2, W3, b3, W4, b4, r);

    const int n4 = out_size / 4;                    // 786432 floats -> 196608 float4
    const int blocks = (n4 + 255) / 256;            // 768 blocks x 256 threads, 1 store each
    broadcast_kernel<<<blocks, 256, 0, stream>>>(r, (float4*)d_out, n4);
}